// FusedMoEWrapper_34918084116584
// MI455X (gfx1250) — compile-verified
//
#include <hip/hip_runtime.h>
#include <math.h>

typedef __attribute__((ext_vector_type(2))) float v2f;
typedef __attribute__((ext_vector_type(8))) float v8f;
typedef __attribute__((ext_vector_type(4))) unsigned tdm_u4;
typedef __attribute__((ext_vector_type(8))) int      tdm_i8;
typedef __attribute__((ext_vector_type(4))) int      tdm_i4;

#define ALPHA_ 1.702f
#define LIMIT_ 7.0f

// Problem sizes fixed by the reference
#define E_     8
#define M_     1024
#define H_     2048
#define I2_    4096
#define INTER_ 2048

#if defined(__has_builtin)
#  if __has_builtin(__builtin_amdgcn_tensor_load_to_lds)
#    define HAVE_TDM 1
#  endif
#endif
#ifndef HAVE_TDM
#  define HAVE_TDM 0
#endif

__device__ __forceinline__ v8f wmma_f32(v2f a, v2f b, v8f c) {
  // D = A(16x4 f32) * B(4x16 f32) + C(16x16 f32)
  return __builtin_amdgcn_wmma_f32_16x16x4_f32(
      /*neg_a=*/false, a, /*neg_b=*/false, b,
      /*c_mod=*/(short)0, c, /*reuse_a=*/false, /*reuse_b=*/false);
}

#if HAVE_TDM
// TDM: DMA a 2D f32 tile (tile_w x tile_h elements, row stride in elements)
// from global memory into LDS, inserting padding after every row so the
// LDS layout is [tile_h][tile_w + pad].
// pad_int_code: pad interval = 2^(code+1) DWORDs; pad_amt_code: pad = code+1 DWORDs.
__device__ __forceinline__ void tdm_load_2d_f32(unsigned lds_byte,
                                                const float* gsrc,
                                                unsigned tile_w, unsigned tile_h,
                                                unsigned long long row_stride_elts,
                                                unsigned pad_int_code,
                                                unsigned pad_amt_code) {
  const unsigned long long ga = (unsigned long long)(size_t)gsrc;
  const unsigned td0 = (unsigned)row_stride_elts;  // tensor width  (>= tile_w)
  const unsigned td1 = 1u << 20;                   // tensor height (large)
  tdm_u4 g0;
  g0[0] = 1u;                                               // count=1 (valid user D#)
  g0[1] = lds_byte;                                         // lds_addr
  g0[2] = (unsigned)ga;                                     // global_addr[31:0]
  g0[3] = (unsigned)((ga >> 32) & 0x01FFFFFFULL) | (2u << 30); // addr[56:32] | type=2
  tdm_i8 g1;
  g1[0] = (int)((2u << 16) |                                // data_size = 4 bytes
                (1u << 20) |                                // pad_enable
                (pad_int_code << 22) | (pad_amt_code << 25));
  g1[1] = (int)((td0 & 0xFFFFu) << 16);                     // tensor_dim0[15:0]
  g1[2] = (int)(((td0 >> 16) & 0xFFFFu) | ((td1 & 0xFFFFu) << 16));
  g1[3] = (int)(((td1 >> 16) & 0xFFFFu) | ((tile_w & 0xFFFFu) << 16)); // tile_dim0
  g1[4] = (int)(tile_h & 0xFFFFu);                          // tile_dim1 (tile_dim2=0)
  g1[5] = (int)(row_stride_elts & 0xFFFFFFFFULL);           // dim0_stride[31:0]
  g1[6] = (int)((row_stride_elts >> 32) & 0xFFFFULL);       // dim0_stride[47:32]
  g1[7] = 0;
  const tdm_i4 z4 = {0, 0, 0, 0};                           // 2D: groups 2/3 unused
  const tdm_i8 z8 = {0, 0, 0, 0, 0, 0, 0, 0};
  __builtin_amdgcn_tensor_load_to_lds(g0, g1, z4, z4, z8, 0);
}
#endif

// ---------------------------------------------------------------------------
// Kernel 1: gu = X @ W1 + b1 (gate/up de-interleaved + K-pair-packed on LDS
// fill), act = (clip(up)+1) * gate*sigmoid(ALPHA*gate) -> workspace.
// Block: 256 threads (8 waves). Tile: 128 rows x 64 inter-cols.
// ---------------------------------------------------------------------------
__global__ __launch_bounds__(256)
void moe_gemm1_act(const float* __restrict__ X, const float* __restrict__ W1,
                   const float* __restrict__ B1, float* __restrict__ Act) {
  const int e  = blockIdx.z;
  const int m0 = blockIdx.y * 128;
  const int n0 = blockIdx.x * 64;    // inter-column base
  const int cb = n0 * 2;             // interleaved gu-column base in W1

  __shared__ float Alds[128][36];    // A tile, padded rows (8/16B aligned)
  __shared__ float Bgp[16][132];     // gate tile, K-pair packed: [k>>1][2n+(k&1)]
  __shared__ float Bup[16][132];     // up   tile, K-pair packed

  const int tid  = threadIdx.x;
  const int wave = tid >> 5;
  const int lane = tid & 31;
  const int half = lane >> 4;
  const int lr   = lane & 15;
  const int wrow = (wave & 3) * 32;
  const int wcol = (wave >> 2) * 32;

  const float* Xe  = X  + (size_t)e * M_ * H_ + (size_t)m0 * H_;
  const float* W1e = W1 + (size_t)e * H_ * I2_;
#if HAVE_TDM
  const unsigned ldsA = (unsigned)(size_t)(void*)&Alds[0][0];
#endif

  v8f cg[2][2] = {};
  v8f cu[2][2] = {};

  for (int k0 = 0; k0 < H_; k0 += 32) {
    __syncthreads();
#if HAVE_TDM
    if (wave == 0)  // one DMA for the whole 128x32 A tile, padded to stride 36
      tdm_load_2d_f32(ldsA, Xe + k0, 32u, 128u, (unsigned long long)H_,
                      /*interval 32 DW*/ 4u, /*pad 4 DW*/ 3u);
#else
#pragma unroll
    for (int i = 0; i < 4; i++) {
      const int lin = tid + i * 256;
      const int r   = lin >> 3;
      const int c4  = lin & 7;
      const float4 v =
          *reinterpret_cast<const float4*>(Xe + (size_t)r * H_ + k0 + c4 * 4);
      Alds[r][c4 * 4 + 0] = v.x; Alds[r][c4 * 4 + 1] = v.y;
      Alds[r][c4 * 4 + 2] = v.z; Alds[r][c4 * 4 + 3] = v.w;
    }
#endif
    // B tile: 32 x 128 interleaved gu columns -> de-interleave into gate/up,
    // K-pair packed so fragments are contiguous 8-byte LDS loads.
#pragma unroll
    for (int i = 0; i < 4; i++) {
      const int lin = tid + i * 256;
      const int r   = lin >> 5;          // k within tile
      const int c   = lin & 31;          // float4 index in row
      const float4 v = *reinterpret_cast<const float4*>(
          W1e + (size_t)(k0 + r) * I2_ + cb + c * 4);
      const int p = r >> 1;
      const int s = r & 1;
      Bgp[p][4 * c + 0 + s] = v.x;   // gate, inter col 2c
      Bup[p][4 * c + 0 + s] = v.y;   // up,   inter col 2c
      Bgp[p][4 * c + 2 + s] = v.z;   // gate, inter col 2c+1
      Bup[p][4 * c + 2 + s] = v.w;   // up,   inter col 2c+1
    }
#if HAVE_TDM
    if (wave == 0) __builtin_amdgcn_s_wait_tensorcnt(0);
#endif
    __syncthreads();

#pragma unroll
    for (int kk = 0; kk < 32; kk += 4) {
      const int p = (kk >> 1) + half;    // K-pair row for this lane-half
      v2f a[2], bg[2], bu[2];
#pragma unroll
      for (int mi = 0; mi < 2; mi++)
        a[mi] = *reinterpret_cast<const v2f*>(
            &Alds[wrow + mi * 16 + lr][kk + 2 * half]);
#pragma unroll
      for (int ni = 0; ni < 2; ni++) {
        const int nb = wcol + ni * 16 + lr;
        bg[ni] = *reinterpret_cast<const v2f*>(&Bgp[p][nb * 2]);
        bu[ni] = *reinterpret_cast<const v2f*>(&Bup[p][nb * 2]);
      }
#pragma unroll
      for (int mi = 0; mi < 2; mi++)
#pragma unroll
        for (int ni = 0; ni < 2; ni++) {
          cg[mi][ni] = wmma_f32(a[mi], bg[ni], cg[mi][ni]);
          cu[mi][ni] = wmma_f32(a[mi], bu[ni], cu[mi][ni]);
        }
    }
  }

  // Epilogue: bias + clamped-SiLU gating, write activated tile
  const float* B1e  = B1 + (size_t)e * I2_;
  float*       Acte = Act + (size_t)e * M_ * INTER_;
#pragma unroll
  for (int mi = 0; mi < 2; mi++) {
#pragma unroll
    for (int ni = 0; ni < 2; ni++) {
      const int n   = n0 + wcol + ni * 16 + lr;
      const float bgv = B1e[2 * n + 0];
      const float buv = B1e[2 * n + 1];
#pragma unroll
      for (int j = 0; j < 8; j++) {
        const int m = m0 + wrow + mi * 16 + j + 8 * half;
        float gate = cg[mi][ni][j] + bgv;
        float up   = cu[mi][ni][j] + buv;
        gate = fminf(gate, LIMIT_);
        up   = fminf(fmaxf(up, -LIMIT_), LIMIT_);
        const float glu = gate / (1.0f + __expf(-ALPHA_ * gate));
        Acte[(size_t)m * INTER_ + n] = (up + 1.0f) * glu;
      }
    }
  }
}

// ---------------------------------------------------------------------------
// Kernel 2: out = Act @ W2 + b2
// Block: 256 threads (8 waves). Tile: 128 x 128. Wave: 32 x 64.
// ---------------------------------------------------------------------------
__global__ __launch_bounds__(256)
void moe_gemm2(const float* __restrict__ Act, const float* __restrict__ W2,
               const float* __restrict__ B2, float* __restrict__ Out) {
  const int e  = blockIdx.z;
  const int m0 = blockIdx.y * 128;
  const int n0 = blockIdx.x * 128;

  __shared__ float Alds[128][36];
  __shared__ float Bp[16][260];      // K-pair packed: [k>>1][2n+(k&1)]

  const int tid  = threadIdx.x;
  const int wave = tid >> 5;
  const int lane = tid & 31;
  const int half = lane >> 4;
  const int lr   = lane & 15;
  const int wrow = (wave & 3) * 32;
  const int wcol = (wave >> 2) * 64;

  const float* Ae = Act + (size_t)e * M_ * INTER_ + (size_t)m0 * INTER_;
  const float* We = W2  + (size_t)e * INTER_ * H_;
#if HAVE_TDM
  const unsigned ldsA = (unsigned)(size_t)(void*)&Alds[0][0];
#endif

  v8f acc[2][4] = {};

  for (int k0 = 0; k0 < INTER_; k0 += 32) {
    __syncthreads();
#if HAVE_TDM
    if (wave == 0)
      tdm_load_2d_f32(ldsA, Ae + k0, 32u, 128u, (unsigned long long)INTER_,
                      4u, 3u);
#else
#pragma unroll
    for (int i = 0; i < 4; i++) {
      const int lin = tid + i * 256;
      const int r   = lin >> 3;
      const int c4  = lin & 7;
      const float4 v = *reinterpret_cast<const float4*>(
          Ae + (size_t)r * INTER_ + k0 + c4 * 4);
      Alds[r][c4 * 4 + 0] = v.x; Alds[r][c4 * 4 + 1] = v.y;
      Alds[r][c4 * 4 + 2] = v.z; Alds[r][c4 * 4 + 3] = v.w;
    }
#endif
#pragma unroll
    for (int i = 0; i < 4; i++) {
      const int lin = tid + i * 256;
      const int r   = lin >> 5;
      const int c   = lin & 31;
      const float4 v = *reinterpret_cast<const float4*>(
          We + (size_t)(k0 + r) * H_ + n0 + c * 4);
      const int p = r >> 1;
      const int s = r & 1;
      Bp[p][8 * c + 0 + s] = v.x;
      Bp[p][8 * c + 2 + s] = v.y;
      Bp[p][8 * c + 4 + s] = v.z;
      Bp[p][8 * c + 6 + s] = v.w;
    }
#if HAVE_TDM
    if (wave == 0) __builtin_amdgcn_s_wait_tensorcnt(0);
#endif
    __syncthreads();

#pragma unroll
    for (int kk = 0; kk < 32; kk += 4) {
      const int p = (kk >> 1) + half;
      v2f a[2], b[4];
#pragma unroll
      for (int mi = 0; mi < 2; mi++)
        a[mi] = *reinterpret_cast<const v2f*>(
            &Alds[wrow + mi * 16 + lr][kk + 2 * half]);
#pragma unroll
      for (int ni = 0; ni < 4; ni++) {
        const int nb = wcol + ni * 16 + lr;
        b[ni] = *reinterpret_cast<const v2f*>(&Bp[p][nb * 2]);
      }
#pragma unroll
      for (int mi = 0; mi < 2; mi++)
#pragma unroll
        for (int ni = 0; ni < 4; ni++)
          acc[mi][ni] = wmma_f32(a[mi], b[ni], acc[mi][ni]);
    }
  }

  const float* B2e  = B2  + (size_t)e * H_;
  float*       Oute = Out + (size_t)e * M_ * H_;
#pragma unroll
  for (int mi = 0; mi < 2; mi++) {
#pragma unroll
    for (int ni = 0; ni < 4; ni++) {
      const int n  = n0 + wcol + ni * 16 + lr;
      const float bv = B2e[n];
#pragma unroll
      for (int j = 0; j < 8; j++) {
        const int m = m0 + wrow + mi * 16 + j + 8 * half;
        Oute[(size_t)m * H_ + n] = acc[mi][ni][j] + bv;
      }
    }
  }
}

extern "C" void kernel_launch(void* const* d_in, const int* in_sizes, int n_in,
                              void* d_out, int out_size, void* d_ws, size_t ws_size,
                              hipStream_t stream) {
  const float* X  = (const float*)d_in[0];  // dispatched (1,1,8,1024,2048)
  const float* W1 = (const float*)d_in[1];  // (8,2048,4096)
  const float* B1 = (const float*)d_in[2];  // (8,4096)
  const float* W2 = (const float*)d_in[3];  // (8,2048,2048)
  const float* B2 = (const float*)d_in[4];  // (8,2048)
  // d_in[5] = sparsity_remap: unused by the reference
  float* Act = (float*)d_ws;                // 8*1024*2048 f32 = 64 MiB
  float* Out = (float*)d_out;               // (8,1,1024,2048) f32

  dim3 g1(INTER_ / 64, M_ / 128, E_);       // (32, 8, 8)
  moe_gemm1_act<<<g1, 256, 0, stream>>>(X, W1, B1, Act);

  dim3 g2(H_ / 128, M_ / 128, E_);          // (16, 8, 8)
  moe_gemm2<<<g2, 256, 0, stream>>>(Act, W2, B2, Out);
}